// Prompt2_49847390437656
// MI455X (gfx1250) — compile-verified
//
#include <hip/hip_runtime.h>
#include <math.h>

// ---------------------------------------------------------------------------
// cos-sim-sum, fully factorized (see reference):
//   s_e[d] = sum_m embed[m,d]         q[d] = sum_m embed[m,d]^2
//   t[d]   = sum_n input[n,d] / max(sqrt(M)*|input[n,d]|, eps)
//   out    = sum_d t[d] * s_e[d] / max(sqrt(q[d]), eps)
// Memory-bound (~96 MB @ 23.3 TB/s ~= 4.1 us). Column sums are done with
// V_WMMA_F32_16X16X4_F32 using an all-ones A matrix: D = 1*B + C makes every
// row of D the column sums of B (invariant to B row permutation).
// ---------------------------------------------------------------------------

typedef float v2f __attribute__((ext_vector_type(2)));
typedef float v8f __attribute__((ext_vector_type(8)));

#define D_DIM            1024
#define EPS_F            1e-8f
#define WAVES_PER_BLOCK  8
#define BLOCK_THREADS    (WAVES_PER_BLOCK * 32)
#define COLS_PER_WAVE    16
#define COLS_PER_BLOCK   (WAVES_PER_BLOCK * COLS_PER_WAVE)   // 128
#define ROWS_PER_CHUNK   128                                 // 32 WMMAs/chain

// MODE 0: write column sum + column sum-of-squares of src chunk.
// MODE 1: write column sum of g(x) = x / max(sqrtM*|x|, eps).
template <int MODE>
__global__ __launch_bounds__(BLOCK_THREADS)
void colreduce_wmma(const float* __restrict__ src,
                    float* __restrict__ psum,
                    float* __restrict__ psq,
                    float sqrtM) {
    const int lane  = threadIdx.x & 31;
    const int wave  = threadIdx.x >> 5;
    const int col   = blockIdx.x * COLS_PER_BLOCK + wave * COLS_PER_WAVE + (lane & 15);
    const int chunk = blockIdx.y;
    // Rows handled by this lane inside each group of 4 (B: 4x16, 2 VGPRs).
    // Exact row->lane mapping is irrelevant under the all-ones A trick; we
    // only need every row of the 4-group covered exactly once per column.
    const int rsub  = (lane >> 4) * 2;            // 0 or 2

    v2f a; a.x = 1.0f; a.y = 1.0f;                // ones A (16x4)
    v8f c  = {};                                  // column-sum accumulator
    v8f c2 = {};                                  // column-sumsq accumulator

    const float* p = src + (size_t)(chunk * ROWS_PER_CHUNK + rsub) * D_DIM + col;

    for (int k = 0; k < ROWS_PER_CHUNK; k += 4) {
        __builtin_prefetch(p + (size_t)ROWS_PER_CHUNK * D_DIM, 0, 0);
        v2f b;
        b.x = p[0];
        b.y = p[D_DIM];
        p += 4 * D_DIM;
        if (MODE == 1) {
            b.x = b.x / fmaxf(sqrtM * fabsf(b.x), EPS_F);
            b.y = b.y / fmaxf(sqrtM * fabsf(b.y), EPS_F);
        }
        c = __builtin_amdgcn_wmma_f32_16x16x4_f32(
                false, a, false, b, (short)0, c, false, false);
        if (MODE == 0) {
            v2f bs; bs.x = b.x * b.x; bs.y = b.y * b.y;
            c2 = __builtin_amdgcn_wmma_f32_16x16x4_f32(
                    false, a, false, bs, (short)0, c2, false, false);
        }
    }

    // Every row of D equals the column sums; D VGPR0 holds N = lane%16.
    if (lane < 16) {
        psum[(size_t)chunk * D_DIM + col] = c[0];
        if (MODE == 0) psq[(size_t)chunk * D_DIM + col] = c2[0];
    }
}

__global__ __launch_bounds__(1024)
void finalize_kernel(const float* __restrict__ psumE,
                     const float* __restrict__ psqE, int chunksE,
                     const float* __restrict__ psumG, int chunksG,
                     float* __restrict__ out) {
    __shared__ float red[1024];
    const int d = threadIdx.x;

    float s = 0.0f, q = 0.0f, t = 0.0f;
    for (int ch = 0; ch < chunksE; ++ch) {
        s += psumE[(size_t)ch * D_DIM + d];
        q += psqE [(size_t)ch * D_DIM + d];
    }
    for (int ch = 0; ch < chunksG; ++ch) {
        t += psumG[(size_t)ch * D_DIM + d];
    }
    const float n2  = sqrtf(q);
    const float val = t * s / fmaxf(n2, EPS_F);

    red[d] = val;
    __syncthreads();
    for (int off = 512; off > 0; off >>= 1) {
        if (d < off) red[d] += red[d + off];
        __syncthreads();
    }
    if (d == 0) out[0] = red[0];
}

extern "C" void kernel_launch(void* const* d_in, const int* in_sizes, int n_in,
                              void* d_out, int out_size, void* d_ws, size_t ws_size,
                              hipStream_t stream) {
    const float* inp = (const float*)d_in[0];   // [N, 1024]
    const float* emb = (const float*)d_in[1];   // [M, 1024]
    const int N = in_sizes[0] / D_DIM;          // 16384
    const int M = in_sizes[1] / D_DIM;          // 8192

    const int chunksE = M / ROWS_PER_CHUNK;     // 64
    const int chunksG = N / ROWS_PER_CHUNK;     // 128

    float* ws    = (float*)d_ws;
    float* psumE = ws;                                        // chunksE * D
    float* psqE  = psumE + (size_t)chunksE * D_DIM;           // chunksE * D
    float* psumG = psqE  + (size_t)chunksE * D_DIM;           // chunksG * D

    dim3 gridE(D_DIM / COLS_PER_BLOCK, chunksE);
    colreduce_wmma<0><<<gridE, BLOCK_THREADS, 0, stream>>>(emb, psumE, psqE, 0.0f);

    dim3 gridG(D_DIM / COLS_PER_BLOCK, chunksG);
    colreduce_wmma<1><<<gridG, BLOCK_THREADS, 0, stream>>>(inp, psumG, nullptr,
                                                           sqrtf((float)M));

    finalize_kernel<<<1, 1024, 0, stream>>>(psumE, psqE, chunksE,
                                            psumG, chunksG, (float*)d_out);
}